// ProductSpaceMessagePassing_57011395887433
// MI455X (gfx1250) — compile-verified
//
#include <hip/hip_runtime.h>
#include <math.h>

typedef __attribute__((ext_vector_type(2))) float v2f;
typedef __attribute__((ext_vector_type(8))) float v8f;

#define DD 64

__device__ __forceinline__ float waveSum(float v) {
#pragma unroll
  for (int off = 16; off > 0; off >>= 1)
    v += __shfl_xor(v, off, 32);
  return v;  // butterfly: every lane ends with the full sum
}

// ---------------------------------------------------------------------------
// Per-row scalar factors:
//   scb[i] = (2/sqrt(c)) * atanh(sqrt(c)*||b_i||) / ||b_i||   (log-map at 0)
//   scs[i] = 1 / max(||s_i||, 1e-12)                          (normalize)
// ---------------------------------------------------------------------------
__global__ void prep_scales_kernel(const float* __restrict__ B,
                                   const float* __restrict__ S,
                                   float* __restrict__ scb,
                                   float* __restrict__ scs,
                                   const float* __restrict__ curv, int n) {
  int lane = threadIdx.x & 31;
  int row  = blockIdx.x * (blockDim.x >> 5) + (threadIdx.x >> 5);
  if (row >= n) return;
  const float* br = B + (size_t)row * DD;
  const float* sr = S + (size_t)row * DD;
  float b0 = br[lane], b1 = br[lane + 32];
  float s0 = sr[lane], s1 = sr[lane + 32];
  float nb = waveSum(b0 * b0 + b1 * b1);
  float ns = waveSum(s0 * s0 + s1 * s1);
  if (lane == 0) {
    float sqc = sqrtf(curv[0]);
    float rb  = fmaxf(sqrtf(nb), 1e-10f);
    float arg = fminf(sqc * rb, 0.9999995f);  // inputs are inside the ball
    scb[row]  = (2.0f / sqc) * atanhf(arg) / rb;
    scs[row]  = 1.0f / fmaxf(sqrtf(ns), 1e-12f);
  }
}

// ---------------------------------------------------------------------------
// H = (rowScale .* X) @ W^T + bias  via V_WMMA_F32_16X16X4_F32.
// One wave -> 16 rows x 64 cols. W (64x64) staged in LDS with pitch 65.
// A frag (16x4 f32): lane L holds row m=L&15, K pair (L>=16 ? k0+2 : k0).
// B frag (4x16 f32): lane L holds col n=L&15, same K-pair selection; B = W^T.
// C/D (16x16 f32): vgpr v -> row v + 8*(L>=16), col L&15.
// ---------------------------------------------------------------------------
__global__ void gemm64_wmma(const float* __restrict__ X,
                            const float* __restrict__ rowScale,
                            const float* __restrict__ W,
                            const float* __restrict__ bias,
                            float* __restrict__ H, int nTiles) {
  __shared__ float sW[DD * 65];
  __shared__ float sB[DD];
  for (int i = threadIdx.x; i < DD * DD; i += blockDim.x)
    sW[(i >> 6) * 65 + (i & 63)] = W[i];
  if (threadIdx.x < DD) sB[threadIdx.x] = bias[threadIdx.x];
  __syncthreads();

  int lane = threadIdx.x & 31;
  int tile = blockIdx.x * (blockDim.x >> 5) + (threadIdx.x >> 5);
  if (tile >= nTiles) return;           // wave-uniform: EXEC stays all-ones
  int r0 = tile << 4;
  int m  = lane & 15;
  int hi = lane >> 4;
  float sc = rowScale ? rowScale[r0 + m] : 1.0f;
  const float* xrow = X + (size_t)(r0 + m) * DD;

  v8f c0 = {}, c1 = {}, c2 = {}, c3 = {};
#pragma unroll
  for (int k0 = 0; k0 < DD; k0 += 4) {
    int ka = k0 + (hi << 1);
    v2f a;
    a.x = xrow[ka] * sc;
    a.y = xrow[ka + 1] * sc;
    v2f b0, b1, b2, b3;
    const float* w0 = &sW[m * 65 + ka];
    const float* w1 = &sW[(m + 16) * 65 + ka];
    const float* w2 = &sW[(m + 32) * 65 + ka];
    const float* w3 = &sW[(m + 48) * 65 + ka];
    b0.x = w0[0]; b0.y = w0[1];
    b1.x = w1[0]; b1.y = w1[1];
    b2.x = w2[0]; b2.y = w2[1];
    b3.x = w3[0]; b3.y = w3[1];
    c0 = __builtin_amdgcn_wmma_f32_16x16x4_f32(false, a, false, b0, (short)0, c0, false, false);
    c1 = __builtin_amdgcn_wmma_f32_16x16x4_f32(false, a, false, b1, (short)0, c1, false, false);
    c2 = __builtin_amdgcn_wmma_f32_16x16x4_f32(false, a, false, b2, (short)0, c2, false, false);
    c3 = __builtin_amdgcn_wmma_f32_16x16x4_f32(false, a, false, b3, (short)0, c3, false, false);
  }

#pragma unroll
  for (int v = 0; v < 8; ++v) {
    float* hrow = H + (size_t)(r0 + v + (hi << 3)) * DD;
    hrow[m]      = c0[v] + sB[m];
    hrow[m + 16] = c1[v] + sB[m + 16];
    hrow[m + 32] = c2[v] + sB[m + 32];
    hrow[m + 48] = c3[v] + sB[m + 48];
  }
}

__global__ void zero_kernel(float* __restrict__ p, size_t n) {
  size_t i      = (size_t)blockIdx.x * blockDim.x + threadIdx.x;
  size_t stride = (size_t)gridDim.x * blockDim.x;
  for (; i < n; i += stride) p[i] = 0.0f;
}

__global__ void normalize_rows_kernel(float* __restrict__ H, int n) {
  int lane = threadIdx.x & 31;
  int row  = blockIdx.x * (blockDim.x >> 5) + (threadIdx.x >> 5);
  if (row >= n) return;
  float* hr = H + (size_t)row * DD;
  float h0 = hr[lane], h1 = hr[lane + 32];
  float inv = 1.0f / fmaxf(sqrtf(waveSum(h0 * h0 + h1 * h1)), 1e-12f);
  hr[lane]      = h0 * inv;
  hr[lane + 32] = h1 * inv;
}

// wave per edge: coalesced 256B row gathers from L2, f32 L2 atomics for scatter
__global__ void scatter_edges_kernel(const float* __restrict__ He,
                                     const float* __restrict__ Hb,
                                     const float* __restrict__ Hs,
                                     const int* __restrict__ src,
                                     const int* __restrict__ dst,
                                     float* __restrict__ ae, float* __restrict__ ab,
                                     float* __restrict__ as_, float* __restrict__ cnt,
                                     int E) {
  int lane = threadIdx.x & 31;
  int e    = blockIdx.x * (blockDim.x >> 5) + (threadIdx.x >> 5);
  if (e >= E) return;
  size_t so   = (size_t)src[e] * DD;
  size_t doff = (size_t)dst[e] * DD;
  atomicAdd(ae  + doff + lane,      He[so + lane]);
  atomicAdd(ae  + doff + lane + 32, He[so + lane + 32]);
  atomicAdd(ab  + doff + lane,      Hb[so + lane]);
  atomicAdd(ab  + doff + lane + 32, Hb[so + lane + 32]);
  atomicAdd(as_ + doff + lane,      Hs[so + lane]);
  atomicAdd(as_ + doff + lane + 32, Hs[so + lane + 32]);
  if (lane == 0) atomicAdd(cnt + (doff >> 6), 1.0f);
}

__global__ void finalize_kernel(const float* __restrict__ ae,
                                const float* __restrict__ ab,
                                const float* __restrict__ as_,
                                const float* __restrict__ cnt,
                                float* __restrict__ oe, float* __restrict__ ob,
                                float* __restrict__ os_,
                                const float* __restrict__ curv, int n) {
  int lane = threadIdx.x & 31;
  int row  = blockIdx.x * (blockDim.x >> 5) + (threadIdx.x >> 5);
  if (row >= n) return;
  size_t off = (size_t)row * DD;
  float inv = 1.0f / fmaxf(cnt[row], 1.0f);

  // euclidean stream: mean -> LeakyReLU(0.2)
  float e0 = ae[off + lane] * inv, e1 = ae[off + lane + 32] * inv;
  oe[off + lane]      = e0 > 0.0f ? e0 : 0.2f * e0;
  oe[off + lane + 32] = e1 > 0.0f ? e1 : 0.2f * e1;

  // hyperbolic stream: mean -> exp-map at origin
  float b0 = ab[off + lane] * inv, b1 = ab[off + lane + 32] * inv;
  float rb  = fmaxf(sqrtf(waveSum(b0 * b0 + b1 * b1)), 1e-10f);
  float sqc = sqrtf(curv[0]);
  float fb  = tanhf(sqc * rb * 0.5f) / (sqc * rb);
  ob[off + lane]      = b0 * fb;
  ob[off + lane + 32] = b1 * fb;

  // spherical stream: mean -> normalize
  float s0 = as_[off + lane] * inv, s1 = as_[off + lane + 32] * inv;
  float is = 1.0f / fmaxf(sqrtf(waveSum(s0 * s0 + s1 * s1)), 1e-12f);
  os_[off + lane]      = s0 * is;
  os_[off + lane + 32] = s1 * is;
}

// ---------------------------------------------------------------------------
extern "C" void kernel_launch(void* const* d_in, const int* in_sizes, int n_in,
                              void* d_out, int out_size, void* d_ws, size_t ws_size,
                              hipStream_t stream) {
  const float* e_emb = (const float*)d_in[0];
  const float* b_emb = (const float*)d_in[1];
  const float* s_emb = (const float*)d_in[2];
  const float* We    = (const float*)d_in[3];
  const float* be    = (const float*)d_in[4];
  const float* Wb    = (const float*)d_in[5];
  const float* bb    = (const float*)d_in[6];
  const float* Ws    = (const float*)d_in[7];
  const float* bs    = (const float*)d_in[8];
  const float* bc    = (const float*)d_in[9];   // b_curvature (device scalar)
  // d_in[10] = s_curvature: unused by the reference forward math
  const int* src = (const int*)d_in[11];
  const int* dst = (const int*)d_in[12];

  const int N = in_sizes[0] / DD;
  const int E = in_sizes[11];
  const int L = in_sizes[3] / (DD * DD);

  size_t nd = (size_t)N * DD;
  float* ws    = (float*)d_ws;
  float* acc_e = ws;            ws += nd;   // accumulators; contiguous with cnt
  float* acc_b = ws;            ws += nd;   // (also serve as layer l-1 outputs)
  float* acc_s = ws;            ws += nd;
  float* cnt   = ws;            ws += N;
  float* He    = ws;            ws += nd;
  float* Hb    = ws;            ws += nd;
  float* Hs    = ws;            ws += nd;
  float* scb   = ws;            ws += N;
  float* scs   = ws;            ws += N;

  float* out_e = (float*)d_out;
  float* out_b = out_e + nd;
  float* out_s = out_b + nd;

  const int rowBlocks  = (N + 7) / 8;          // 8 waves/block, wave per row
  const int nTiles     = (N + 15) / 16;        // N=50000 -> exactly 3125 tiles
  const int gemmBlocks = (nTiles + 3) / 4;     // 4 waves/block
  const int edgeBlocks = (E + 7) / 8;          // wave per edge
  const size_t zeroN   = 3 * nd + (size_t)N;   // acc_e|acc_b|acc_s|cnt contiguous

  for (int l = 0; l < L; ++l) {
    const float* Xe = (l == 0) ? e_emb : acc_e;
    const float* Xb = (l == 0) ? b_emb : acc_b;
    const float* Xs = (l == 0) ? s_emb : acc_s;
    float* Oe = (l == L - 1) ? out_e : acc_e;
    float* Ob = (l == L - 1) ? out_b : acc_b;
    float* Os = (l == L - 1) ? out_s : acc_s;

    prep_scales_kernel<<<rowBlocks, 256, 0, stream>>>(Xb, Xs, scb, scs, bc, N);
    gemm64_wmma<<<gemmBlocks, 128, 0, stream>>>(Xe, nullptr, We + (size_t)l * DD * DD,
                                                be + (size_t)l * DD, He, nTiles);
    gemm64_wmma<<<gemmBlocks, 128, 0, stream>>>(Xb, scb, Wb + (size_t)l * DD * DD,
                                                bb + (size_t)l * DD, Hb, nTiles);
    gemm64_wmma<<<gemmBlocks, 128, 0, stream>>>(Xs, scs, Ws + (size_t)l * DD * DD,
                                                bs + (size_t)l * DD, Hs, nTiles);
    // acc buffers (== layer inputs for l>0) are dead once the GEMMs have read them
    zero_kernel<<<2048, 256, 0, stream>>>(acc_e, zeroN);
    normalize_rows_kernel<<<rowBlocks, 256, 0, stream>>>(Hs, N);
    scatter_edges_kernel<<<edgeBlocks, 256, 0, stream>>>(He, Hb, Hs, src, dst,
                                                         acc_e, acc_b, acc_s, cnt, E);
    finalize_kernel<<<rowBlocks, 256, 0, stream>>>(acc_e, acc_b, acc_s, cnt,
                                                   Oe, Ob, Os, bc, N);
  }
}